// SPP_6536940225042
// MI455X (gfx1250) — compile-verified
//
#include <hip/hip_runtime.h>
#include <cstdint>

// SPP / SPPF for x = [16, 256, 64, 64] f32.
// out = concat(x, pool5(x), pool9(x), pool13(x)) on channel axis.
// pool9 = pool5(pool5(x)), pool13 = pool5(pool9) exactly (−inf padding),
// and each pool5 is separable (radius-2 H pass, then radius-2 V pass).
//
// One workgroup per (b,c) plane. Plane staged into LDS via CDNA5 async
// global->LDS copies (ASYNCcnt path); all pooling reads come from LDS;
// all 4 output sections written with non-temporal stores (pure streaming).

typedef float f32x4 __attribute__((ext_vector_type(4)));

#define HH 64
#define WW 64
#define PR 2                   // row halo (radius 2)
#define PC 4                   // col halo (4 for aligned f32x4 halo loads)
#define LR (HH + 2 * PR)       // 68 padded rows
#define LC (WW + 2 * PC)       // 72 floats row stride (16B-aligned quads)
#define NTHREADS 256

// LDS byte offset (within the workgroup's LDS block) of a __shared__ pointer.
__device__ __forceinline__ unsigned lds_off(const void* p) {
  return (unsigned)(unsigned long long)(__attribute__((address_space(3))) const char*)p;
}

// CDNA5 async copy: 16 bytes per lane, global -> LDS, tracked by ASYNCcnt.
__device__ __forceinline__ void async_copy_b128(unsigned lds_byte,
                                                unsigned long long gaddr) {
  asm volatile("global_load_async_to_lds_b128 %0, %1, off"
               :: "v"(lds_byte), "v"(gaddr)
               : "memory");
}

__device__ __forceinline__ void wait_async0() {
  asm volatile("s_wait_asynccnt 0" ::: "memory");
}

__device__ __forceinline__ f32x4 vmax4(f32x4 a, f32x4 b) {
  f32x4 r;
  r.x = fmaxf(a.x, b.x);
  r.y = fmaxf(a.y, b.y);
  r.z = fmaxf(a.z, b.z);
  r.w = fmaxf(a.w, b.w);
  return r;
}

__global__ __launch_bounds__(NTHREADS) void spp_kernel(
    const float* __restrict__ x, float* __restrict__ out) {
  __shared__ float A[LR * LC];   // current stage input (x, then m5, then m9)
  __shared__ float T[LR * LC];   // horizontal-pass temporary

  const int tid   = threadIdx.x;
  const int plane = blockIdx.x;          // b*256 + c
  const int b     = plane >> 8;
  const int c     = plane & 255;
  const float NEG = -__builtin_inff();

  const float* xp = x + (size_t)plane * (HH * WW);
  float* op = out + ((size_t)b * 1024 + (size_t)c) * (size_t)(HH * WW);
  const size_t sec = (size_t)256 * HH * WW;   // channel-section stride

  // Thread -> 4 rows x 4 contiguous cols. tid covers rows 0..15 once;
  // j adds 16 rows. Columns are 16B aligned.
  const int r0 = tid >> 4;                 // 0..15
  const int c0 = (tid & 15) << 2;          // 0,4,...,60

  // 1) Issue async global->LDS copies of the plane's data region.
#pragma unroll
  for (int j = 0; j < 4; ++j) {
    const int r = r0 + 16 * j;
    async_copy_b128(lds_off(&A[(PR + r) * LC + PC + c0]),
                    (unsigned long long)(uintptr_t)(xp + r * WW + c0));
  }

  // 2) Fill -inf halos while the async copies are in flight.
  //    A: only halo cells (data region is being written by the ASYNCcnt-path
  //    async loads). T: everything (data region is rewritten each H pass,
  //    row halos must stay -inf for the V pass).
  for (int i = tid; i < LR * LC; i += NTHREADS) {
    const int rr = i / LC;
    const int cc = i - rr * LC;
    if (rr < PR || rr >= PR + HH || cc < PC || cc >= PC + WW) A[i] = NEG;
    T[i] = NEG;
  }

  wait_async0();
  __syncthreads();

  // 3) Section 0: pass-through copy of x (streaming NT stores).
#pragma unroll
  for (int j = 0; j < 4; ++j) {
    const int r = r0 + 16 * j;
    f32x4 v = *(const f32x4*)&A[(PR + r) * LC + PC + c0];
    __builtin_nontemporal_store(v, (f32x4*)(op + r * WW + c0));
  }

  // 4) SPPF cascade: three radius-2 separable max pools.
  for (int s = 1; s <= 3; ++s) {
    // Horizontal radius-2 max: A -> T (data rows only; col halo is -inf).
#pragma unroll
    for (int j = 0; j < 4; ++j) {
      const int r = r0 + 16 * j;
      const float* row = &A[(PR + r) * LC + PC + c0];
      f32x4 a  = *(const f32x4*)(row - 4);   // cols c0-4 .. c0-1
      f32x4 bm = *(const f32x4*)(row);       // cols c0   .. c0+3
      f32x4 cm = *(const f32x4*)(row + 4);   // cols c0+4 .. c0+7
      const float m_xyz = fmaxf(fmaxf(bm.x, bm.y), bm.z);
      const float m_yzw = fmaxf(fmaxf(bm.y, bm.z), bm.w);
      const float m_all = fmaxf(m_xyz, bm.w);
      f32x4 o;
      o.x = fmaxf(a.z, fmaxf(a.w, m_xyz));          // max cols c0-2..c0+2
      o.y = fmaxf(a.w, m_all);                      // max cols c0-1..c0+3
      o.z = fmaxf(m_all, cm.x);                     // max cols c0  ..c0+4
      o.w = fmaxf(m_yzw, fmaxf(cm.x, cm.y));        // max cols c0+1..c0+5
      *(f32x4*)&T[(PR + r) * LC + PC + c0] = o;
    }
    __syncthreads();

    // Vertical radius-2 max: T -> A, stream result to output section s.
#pragma unroll
    for (int j = 0; j < 4; ++j) {
      const int r = r0 + 16 * j;
      const float* colp = &T[r * LC + PC + c0];     // padded rows r .. r+4
      f32x4 v = *(const f32x4*)(colp);
      v = vmax4(v, *(const f32x4*)(colp + 1 * LC));
      v = vmax4(v, *(const f32x4*)(colp + 2 * LC));
      v = vmax4(v, *(const f32x4*)(colp + 3 * LC));
      v = vmax4(v, *(const f32x4*)(colp + 4 * LC));
      *(f32x4*)&A[(PR + r) * LC + PC + c0] = v;     // input for next stage
      __builtin_nontemporal_store(v, (f32x4*)(op + s * sec + r * WW + c0));
    }
    __syncthreads();
  }
}

extern "C" void kernel_launch(void* const* d_in, const int* in_sizes, int n_in,
                              void* d_out, int out_size, void* d_ws,
                              size_t ws_size, hipStream_t stream) {
  const float* x = (const float*)d_in[0];
  float* out = (float*)d_out;
  const int planes = in_sizes[0] / (HH * WW);   // 16 * 256 = 4096
  spp_kernel<<<dim3(planes), dim3(NTHREADS), 0, stream>>>(x, out);
}